// DialogueGCNC_71038759076671
// MI455X (gfx1250) — compile-verified
//
#include <hip/hip_runtime.h>
#include <cstddef>

// Problem constants (match reference)
#define N_UTT  16384
#define D_DIM  256
#define W_HALF 64
#define TW     128      // 2*W
#define JSPAN  144      // 16 rows + 128 window
#define JPAD   164      // padded LDS width: row stride 164 -> 16 distinct banks
#define BPAD   260      // padded buf width: avoids half-wave store conflicts

typedef __attribute__((ext_vector_type(2))) float v2f;
typedef __attribute__((ext_vector_type(8))) float v8f;

// fp32 WMMA: D(16x16,f32) = A(16x4,f32) * B(4x16,f32) + C
// A layout: lanes 0-15 -> M=lane, {v0,v1}={K,K+1}; lanes 16-31 -> M=lane-16, {K+2,K+3}
// B layout: lanes 0-15 -> {v0,v1}=rows{K,K+1}; lanes 16-31 -> rows{K+2,K+3}, col=lane&15
// C layout: vgpr g -> M = g + 8*(lane>=16), col = lane&15
__device__ __forceinline__ v8f wmma4_f32(v2f a, v2f b, v8f c) {
#if defined(__gfx1250__)
  return __builtin_amdgcn_wmma_f32_16x16x4_f32(
      /*neg_a=*/false, a, /*neg_b=*/false, b,
      /*c_mod=*/(short)0, c, /*reuse_a=*/false, /*reuse_b=*/false);
#else
  (void)a; (void)b; return c;  // host-pass parse only
#endif
}

// ---------------------------------------------------------------------------
// Kernel 1: Y_g = x @ W_g for g in {att,pred,suc,same,diff} -> ws[g*N*D ...]
// grid = (N/16, 5), block = 256 (8 waves). Wave w computes cols [32w, 32w+32).
// Pointer select via ternary chain keeps global address space (no flat loads).
// ---------------------------------------------------------------------------
__global__ __launch_bounds__(256) void gemm5_kernel(
    const float* __restrict__ x,
    const float* __restrict__ Watt, const float* __restrict__ Wpred,
    const float* __restrict__ Wsuc, const float* __restrict__ Wsame,
    const float* __restrict__ Wdiff, float* __restrict__ ws)
{
  const int i0 = blockIdx.x * 16;
  const int g  = blockIdx.y;
  const float* Wm = (g == 0) ? Watt
                  : (g == 1) ? Wpred
                  : (g == 2) ? Wsuc
                  : (g == 3) ? Wsame
                             : Wdiff;
  float* Y = ws + (size_t)g * N_UTT * D_DIM;

  const int lane = threadIdx.x & 31;
  const int wid  = threadIdx.x >> 5;
  const int half = lane >> 4;     // 0: K,K+1   1: K+2,K+3
  const int l16  = lane & 15;

  const float* aptr = x + (size_t)(i0 + l16) * D_DIM + 2 * half;
  const int c0 = wid * 32;
  const int c1 = c0 + 16;
  const float* b0 = Wm + (size_t)(2 * half) * D_DIM + c0 + l16;
  const float* b1 = Wm + (size_t)(2 * half) * D_DIM + c1 + l16;

  // warm L2 for the A row and the first W cachelines this wave streams
  __builtin_prefetch(aptr, 0, 1);
  __builtin_prefetch(b0, 0, 1);
  __builtin_prefetch(b1, 0, 1);

  v8f acc0 = {}; v8f acc1 = {};
  for (int k = 0; k < D_DIM; k += 4) {
    v2f a = *(const v2f*)(aptr + k);
    v2f bb0, bb1;
    bb0.x = b0[(size_t)k       * D_DIM];
    bb0.y = b0[(size_t)(k + 1) * D_DIM];
    bb1.x = b1[(size_t)k       * D_DIM];
    bb1.y = b1[(size_t)(k + 1) * D_DIM];
    acc0 = wmma4_f32(a, bb0, acc0);
    acc1 = wmma4_f32(a, bb1, acc1);
  }
  for (int gg = 0; gg < 8; ++gg) {
    int row = i0 + gg + 8 * half;
    Y[(size_t)row * D_DIM + c0 + l16] = acc0[gg];
    Y[(size_t)row * D_DIM + c1 + l16] = acc1[gg];
  }
}

// ---------------------------------------------------------------------------
// Kernel 2: fused banded attention + relation GCN aggregation + log_softmax.
// One block (8 waves) per 16-row tile.
// ---------------------------------------------------------------------------
__global__ __launch_bounds__(256) void fused_attn_agg_kernel(
    const float* __restrict__ x, const int* __restrict__ spk,
    const float* __restrict__ ws, float* __restrict__ out)
{
  __shared__ float coeff[4][16][JPAD];  // masked-attn coefficient matrices
  __shared__ float buf[16][BPAD];       // raw[16][144] in A/B, h[16][256] in C/D

  const int i0   = blockIdx.x * 16;
  const int tid  = threadIdx.x;
  const int lane = tid & 31;
  const int wid  = tid >> 5;
  const int half = lane >> 4;
  const int l16  = lane & 15;

  const float* q = ws;  // group 0

  // zero-init coefficient matrices (entries off the band are never written)
  for (int idx = tid; idx < 4 * 16 * JPAD; idx += 256)
    ((float*)coeff)[idx] = 0.0f;

  // warm L2 for the 4 support panels this block will read in Phase C
  {
    const int jc = min(max(i0 - W_HALF + (int)tid, 0), N_UTT - 1);
    for (int rel = 0; rel < 4; ++rel)
      __builtin_prefetch(ws + (size_t)(rel + 1) * N_UTT * D_DIM
                             + (size_t)jc * D_DIM, 0, 1);
  }

  // ---- Phase A: raw[r][j'] = q[i0+r] . x[i0-64+j'],  j' in [0,144) ----
  {
    const float* aptr = q + (size_t)(i0 + l16) * D_DIM + 2 * half;
    for (int t = wid; t < 9; t += 8) {           // 9 column tiles of 16
      const int jp = t * 16 + l16;
      const int j  = i0 - W_HALF + jp;
      const bool valid = (j >= 0) && (j < N_UTT);
      const float* bptr = x + (size_t)(valid ? j : 0) * D_DIM + 2 * half;
      v8f acc = {};
      for (int k = 0; k < D_DIM; k += 4) {
        v2f a = *(const v2f*)(aptr + k);
        v2f b;
        if (valid) { b = *(const v2f*)(bptr + k); }
        else       { b.x = 0.0f; b.y = 0.0f; }   // zero-pad -> raw contribution 0
        acc = wmma4_f32(a, b, acc);
      }
      for (int gg = 0; gg < 8; ++gg)
        buf[gg + 8 * half][t * 16 + l16] = acc[gg];
    }
  }
  __syncthreads();

  // ---- Phase B: per-row softmax over 128 band slots + build coeffs ----
  for (int rr = 0; rr < 2; ++rr) {
    const int r = wid * 2 + rr;
    const int i = i0 + r;
    const int spk_i = spk[i];
    float v[4];
    for (int kk = 0; kk < 4; ++kk) {
      const int w = lane + 32 * kk;
      v[kk] = buf[r][r + w];
    }
    float m = fmaxf(fmaxf(v[0], v[1]), fmaxf(v[2], v[3]));
    for (int o = 16; o > 0; o >>= 1) m = fmaxf(m, __shfl_xor(m, o, 32));
    float e[4]; float s = 0.0f;
    for (int kk = 0; kk < 4; ++kk) { e[kk] = expf(v[kk] - m); s += e[kk]; }
    for (int o = 16; o > 0; o >>= 1) s += __shfl_xor(s, o, 32);
    const float inv = 1.0f / s;
    for (int kk = 0; kk < 4; ++kk) {
      const int w   = lane + 32 * kk;
      const int off = w - W_HALF;
      const int j   = i + off;
      const bool valid = (j >= 0) && (j < N_UTT);
      const float a = e[kk] * inv;
      const int jp  = r + w;
      const int sj  = valid ? spk[j] : -1;
      coeff[0][r][jp] = (valid && off >= 0)     ? a : 0.0f;  // pred
      coeff[1][r][jp] = (valid && off <  0)     ? a : 0.0f;  // suc
      coeff[2][r][jp] = (valid && sj == spk_i)  ? a : 0.0f;  // same
      coeff[3][r][jp] = (valid && sj != spk_i)  ? a : 0.0f;  // diff
    }
  }
  __syncthreads();

  // ---- Phase C: h = sum_r coeff_r[16x144] @ S_r[144x256] ----
  for (int tb = 0; tb < 2; ++tb) {
    const int c0 = (wid + 8 * tb) * 16;
    v8f acc = {};
    for (int rel = 0; rel < 4; ++rel) {
      const float* S = ws + (size_t)(rel + 1) * N_UTT * D_DIM;
      for (int k = 0; k < JSPAN; k += 4) {
        const int kp = k + 2 * half;
        v2f a;
        a.x = coeff[rel][l16][kp];
        a.y = coeff[rel][l16][kp + 1];
        const int j0  = i0 - W_HALF + kp;       // clamp addr; coeff=0 kills OOB
        const int jr0 = min(max(j0, 0), N_UTT - 1);
        const int jr1 = min(max(j0 + 1, 0), N_UTT - 1);
        v2f b;
        b.x = S[(size_t)jr0 * D_DIM + c0 + l16];
        b.y = S[(size_t)jr1 * D_DIM + c0 + l16];
        acc = wmma4_f32(a, b, acc);
      }
    }
    for (int gg = 0; gg < 8; ++gg)
      buf[gg + 8 * half][c0 + l16] = acc[gg];
  }
  __syncthreads();

  // ---- Phase D: out = log_softmax(h, axis=1) ----
  for (int rr = 0; rr < 2; ++rr) {
    const int r = wid * 2 + rr;
    float v[8]; float m = -1e30f;
    for (int kk = 0; kk < 8; ++kk) {
      v[kk] = buf[r][lane + 32 * kk];
      m = fmaxf(m, v[kk]);
    }
    for (int o = 16; o > 0; o >>= 1) m = fmaxf(m, __shfl_xor(m, o, 32));
    float s = 0.0f;
    for (int kk = 0; kk < 8; ++kk) s += expf(v[kk] - m);
    for (int o = 16; o > 0; o >>= 1) s += __shfl_xor(s, o, 32);
    const float L = m + logf(s);
    float* orow = out + (size_t)(i0 + r) * D_DIM;
    for (int kk = 0; kk < 8; ++kk) orow[lane + 32 * kk] = v[kk] - L;
  }
}

// ---------------------------------------------------------------------------
extern "C" void kernel_launch(void* const* d_in, const int* in_sizes, int n_in,
                              void* d_out, int out_size, void* d_ws, size_t ws_size,
                              hipStream_t stream) {
  (void)in_sizes; (void)n_in; (void)out_size; (void)ws_size;
  const float* x     = (const float*)d_in[0];
  const int*   spk   = (const int*)d_in[1];
  const float* Watt  = (const float*)d_in[2];
  const float* Wpred = (const float*)d_in[3];
  const float* Wsuc  = (const float*)d_in[4];
  const float* Wsame = (const float*)d_in[5];
  const float* Wdiff = (const float*)d_in[6];
  float* ws  = (float*)d_ws;   // needs 5*N*D*4 = 80 MB: q + 4 supports
  float* out = (float*)d_out;

  dim3 g1(N_UTT / 16, 5);
  gemm5_kernel<<<g1, 256, 0, stream>>>(x, Watt, Wpred, Wsuc, Wsame, Wdiff, ws);
  fused_attn_agg_kernel<<<N_UTT / 16, 256, 0, stream>>>(x, spk, ws, out);
}